// EMAVectorQuantizer_72559177498959
// MI455X (gfx1250) — compile-verified
//
#include <hip/hip_runtime.h>
#include <hip/hip_bf16.h>

typedef float v2f __attribute__((ext_vector_type(2)));
typedef float v8f __attribute__((ext_vector_type(8)));

#define NROWS   131072      // 32*64*64
#define DIM     256
#define NCODE   1024
#define MTILE   32          // rows per workgroup
#define NBLOCKS (NROWS / MTILE)              // 4096
#define QELEMS  (NROWS * (size_t)DIM)        // 33554432
#define APITCH  260         // 256+4 pad: 1040B row = 65*16 (b128-aligned, banks spread)
#define BPITCH  36          // 32+4 pad:  144B row = 9*16  (b128-aligned, banks spread)
#define KBLK    32          // K per pipeline stage
#define NSTAGE  (NCODE / 64 * (DIM / KBLK))  // 16 chunks * 8 kblocks = 128

// CDNA5 async global->LDS copy (ASYNCcnt-tracked, no VGPR round trip).
// Generic pointers to __shared__ carry the LDS byte offset in their low 32 bits.
__device__ __forceinline__ void async_b128(unsigned long long gbase,
                                           unsigned byte_off, void* ldsdst)
{
    unsigned lds = (unsigned)(unsigned long long)ldsdst;
    asm volatile("global_load_async_to_lds_b128 %0, %1, %2"
                 :: "v"(lds), "v"(byte_off), "s"(gbase) : "memory");
}

// ---------------------------------------------------------------------------
// Kernel 1: fused scores-GEMM (f32 WMMA) + per-row argmax + loss partials.
// 256 thr = 8 waves (2 M-subtiles x 4 N-subtiles); double-buffered async B.
// ---------------------------------------------------------------------------
__global__ __launch_bounds__(256)
void vq_score(const float* __restrict__ in, const float* __restrict__ cb,
              int* __restrict__ ws_idx, float* __restrict__ partials)
{
    __shared__ __align__(16) float Ash[MTILE][APITCH];   // 33,280 B
    __shared__ __align__(16) float Bsh[2][64][BPITCH];   // 18,432 B
    __shared__ float redS[2][4][16];
    __shared__ int   redI[2][4][16];
    __shared__ float rownorm[MTILE];
    __shared__ float lossP[MTILE];

    const int tid  = threadIdx.x;
    const int wave = tid >> 5;
    const int lane = tid & 31;
    const int lr   = lane & 15;       // row-within-16 / code-within-16
    const int lh   = lane >> 4;       // half-wave select
    const int m    = wave >> 2;       // 0..1
    const int n    = wave & 3;        // 0..3

    // ---- stage A asynchronously: 32 rows x 256 f32 (8 b128s in flight) ----
    const unsigned long long inbase =
        (unsigned long long)(in + (size_t)blockIdx.x * (MTILE * DIM));
    #pragma unroll
    for (int i = 0; i < 8; ++i) {
        int g4  = tid + 256 * i;            // 0..2047
        int row = g4 >> 6;                  // 64 float4 per row
        int col = (g4 & 63) << 2;
        async_b128(inbase, (unsigned)g4 * 16u, &Ash[row][col]);
    }
    asm volatile("s_wait_asynccnt 0x0" ::: "memory");
    __syncthreads();

    // ---- per-row ||x||^2 (tiny) ----
    if (tid < MTILE) {
        float s = 0.f;
        #pragma unroll 8
        for (int k = 0; k < DIM; ++k) { float x = Ash[tid][k]; s += x * x; }
        rownorm[tid] = s;
    }

    float best[8];
    int   bidx[8];
    #pragma unroll
    for (int r = 0; r < 8; ++r) { best[r] = -3.4e38f; bidx[r] = 0; }

    const unsigned long long cbbase = (unsigned long long)cb;

    // stage s covers codes [ (s>>3)*64, +64 ) x k [ (s&7)*32, +32 )
    auto issueB = [&](int stage, int bsel) {
        int chunk = stage >> 3, kb = stage & 7;
        #pragma unroll
        for (int i = 0; i < 2; ++i) {
            int g4   = tid + 256 * i;                  // 0..511
            int cl   = g4 >> 3;                        // code-local (8 f4/row)
            int col4 = g4 & 7;
            unsigned goff = (unsigned)(((chunk * 64 + cl) * 64 + kb * 8 + col4) * 16);
            async_b128(cbbase, goff, &Bsh[bsel][cl][col4 << 2]);
        }
    };

    issueB(0, 0);
    int buf = 0;
    const int arow = 16 * m + lr;
    const int brow = 16 * n + lr;

    #pragma unroll 1
    for (int chunk = 0; chunk < NCODE / 64; ++chunk) {
        v8f acc = {0.f, 0.f, 0.f, 0.f, 0.f, 0.f, 0.f, 0.f};

        #pragma unroll 1
        for (int kb = 0; kb < DIM / KBLK; ++kb) {
            int stage = chunk * (DIM / KBLK) + kb;
            if (stage < NSTAGE - 1) {
                issueB(stage + 1, buf ^ 1);            // prefetch next tile
                asm volatile("s_wait_asynccnt 0x2" ::: "memory"); // this tile done
            } else {
                asm volatile("s_wait_asynccnt 0x0" ::: "memory");
            }
            __syncthreads();

            #pragma unroll
            for (int kk = 0; kk < KBLK / 4; ++kk) {
                // ISA fragment layout: float2 at [row=lane%16][k0 + 2*(lane/16)]
                v2f a = *(const v2f*)&Ash[arow][kb * KBLK + kk * 4 + 2 * lh];
                v2f b = *(const v2f*)&Bsh[buf][brow][kk * 4 + 2 * lh];
                acc = __builtin_amdgcn_wmma_f32_16x16x4_f32(
                        false, a, false, b, (short)0, acc, false, false);
            }
            __syncthreads();                            // all reads of buf done
            buf ^= 1;
        }

        // fold this chunk's 16x16 score tile into per-lane running best
        const int code = chunk * 64 + 16 * n + lr;
        #pragma unroll
        for (int r = 0; r < 8; ++r) {
            float s = acc[r];
            if (s > best[r]) { best[r] = s; bidx[r] = code; }
        }
    }

    // ---- cross-lane argmax within each 16-lane half (C tile N dimension) ----
    #pragma unroll
    for (int r = 0; r < 8; ++r) {
        float s = best[r];
        int   c = bidx[r];
        #pragma unroll
        for (int off = 8; off >= 1; off >>= 1) {
            float so = __shfl_xor(s, off, 16);
            int   co = __shfl_xor(c, off, 16);
            if (so > s || (so == s && co < c)) { s = so; c = co; }
        }
        if (lr == 0) {                       // lane 0 -> rows r, lane 16 -> rows r+8
            redS[m][n][r + 8 * lh] = s;
            redI[m][n][r + 8 * lh] = c;
        }
    }
    __syncthreads();

    // ---- cross-wave merge (4 N-subtile waves per M row), idx + loss ----
    if (tid < MTILE) {
        int mm = tid >> 4, rr = tid & 15;
        float bs = redS[mm][0][rr];
        int   bi = redI[mm][0][rr];
        #pragma unroll
        for (int nn = 1; nn < 4; ++nn) {
            float s2 = redS[mm][nn][rr];
            int   i2 = redI[mm][nn][rr];
            if (s2 > bs || (s2 == bs && i2 < bi)) { bs = s2; bi = i2; }
        }
        ws_idx[blockIdx.x * MTILE + tid] = bi;
        lossP[tid] = rownorm[tid] - 2.0f * bs;   // ||x-e||^2 for this row
    }
    __syncthreads();
    if (tid == 0) {
        float t = 0.f;
        #pragma unroll
        for (int i = 0; i < MTILE; ++i) t += lossP[i];
        partials[blockIdx.x] = t;
    }
}

// ---------------------------------------------------------------------------
// Kernel 2: quantized = codebook[idx] (coalesced, codebook L2-resident),
//           plus idx written as float into the output tail
// ---------------------------------------------------------------------------
__global__ __launch_bounds__(256)
void vq_writeout(const float* __restrict__ cb, const int* __restrict__ ws_idx,
                 float* __restrict__ qout, float* __restrict__ idxout)
{
    int f4  = blockIdx.x * 256 + threadIdx.x;   // 0 .. 8388607
    int row = f4 >> 6;                          // 64 float4 per row
    int c4  = f4 & 63;
    int code = ws_idx[row];
    float4 v = ((const float4*)cb)[code * (DIM / 4) + c4];
    ((float4*)qout)[f4] = v;
    if (c4 == 0) idxout[row] = (float)code;
}

// ---------------------------------------------------------------------------
// Kernel 3: deterministic loss reduction over per-block partials
// ---------------------------------------------------------------------------
__global__ __launch_bounds__(256)
void vq_loss(const float* __restrict__ partials, float* __restrict__ lossout)
{
    __shared__ float sm[256];
    float s = 0.f;
    for (int i = threadIdx.x; i < NBLOCKS; i += 256) s += partials[i];
    sm[threadIdx.x] = s;
    __syncthreads();
    for (int off = 128; off >= 1; off >>= 1) {
        if (threadIdx.x < off) sm[threadIdx.x] += sm[threadIdx.x + off];
        __syncthreads();
    }
    if (threadIdx.x == 0)
        lossout[0] = 1.25f * sm[0] / (float)QELEMS;   // q_loss + BETA*e_loss
}

extern "C" void kernel_launch(void* const* d_in, const int* in_sizes, int n_in,
                              void* d_out, int out_size, void* d_ws, size_t ws_size,
                              hipStream_t stream)
{
    const float* in = (const float*)d_in[0];   // [32,256,64,64] f32
    const float* cb = (const float*)d_in[1];   // [1024,256]     f32

    int*   ws_idx   = (int*)d_ws;                                   // 131072 ints
    float* partials = (float*)((char*)d_ws + NROWS * sizeof(int));  // 4096 floats

    float* qout    = (float*)d_out;            // [0, 33554432) quantized_st
    float* lossout = qout + QELEMS;            // [33554432]    loss scalar
    float* idxout  = qout + QELEMS + 1;        // [+1, +131072] idx as float

    vq_score<<<NBLOCKS, 256, 0, stream>>>(in, cb, ws_idx, partials);
    vq_writeout<<<(int)(QELEMS / 4 / 256), 256, 0, stream>>>(cb, ws_idx, qout, idxout);
    vq_loss<<<1, 256, 0, stream>>>(partials, lossout);
}